// GCN_1202590843048
// MI455X (gfx1250) — compile-verified
//
#include <hip/hip_runtime.h>
#include <hip/hip_bf16.h>

#define DCH   128
#define PSTR  288           // LDS stride (floats) per K-pair row: 288 % 64 == 32
                            // -> the two 16-lane halves of a wave read disjoint banks
#define BN_EPS 1e-5f

typedef __attribute__((ext_vector_type(2))) float v2f;
typedef __attribute__((ext_vector_type(8))) float v8f;

// ---------------------------------------------------------------------------
// GEMM: H[n,128] = pre(A)[n,128] @ W[128,128] using V_WMMA_F32_16X16X4_F32.
// pre(a) = PRE ? max(a*preScale[k] + preShift[k], 0) : a   (fused BN+bias+ReLU
// of the *previous* layer, applied on load).
//
// W is staged into LDS K-paired: element (k,n) -> Ws[(k>>1)*PSTR + n*2 + (k&1)],
// so a B fragment (W[ka][n], W[ka+1][n]) is one aligned ds_load_b64 directly
// into a legal even-aligned WMMA operand pair (kills the v_dual_mov shuffles).
//
// Block = 256 threads (8 wave32s). Each wave computes a 16-row strip x 128
// cols as 8 accumulator tiles; the A fragment is reused across all 8 tiles.
// ---------------------------------------------------------------------------
template <bool PRE>
__global__ __launch_bounds__(256)
void gemm_wmma(const float* __restrict__ A, const float* __restrict__ W,
               float* __restrict__ H, int nrows,
               const float* __restrict__ preScale,
               const float* __restrict__ preShift)
{
    __shared__ float Ws[(DCH / 2) * PSTR];   // 64 pair-rows * 288 floats = 72KB

    const int tid = threadIdx.x;
    // Cooperative load of W (128x128 f32): float4 from global, K-paired scatter
    // into LDS (one-time cost).
    for (int idx = tid; idx < DCH * (DCH / 4); idx += 256) {
        const int r  = idx >> 5;          // row (K) 0..127
        const int c4 = (idx & 31) << 2;   // col (N) 0..124 step 4
        const float4 w4 = *reinterpret_cast<const float4*>(W + r * DCH + c4);
        float* dstp = Ws + (r >> 1) * PSTR + (r & 1);
        dstp[(c4 + 0) * 2] = w4.x;
        dstp[(c4 + 1) * 2] = w4.y;
        dstp[(c4 + 2) * 2] = w4.z;
        dstp[(c4 + 3) * 2] = w4.w;
    }
    __syncthreads();

    const int lane  = tid & 31;
    const int wave  = tid >> 5;        // 0..7
    const int mhalf = lane >> 4;       // 0|1 (K-half for A/B frags)
    const int ml    = lane & 15;       // M (A) / N (B) position
    const int rowBase = blockIdx.x * 128 + wave * 16;

    // Clamp A row so no lane diverges before the WMMA region (EXEC all-ones).
    int arow = rowBase + ml;
    if (arow > nrows - 1) arow = nrows - 1;
    const float* __restrict__ Arow = A + (size_t)arow * DCH;

    // Per-lane LDS base for B fragments: pair-row (mhalf) + column (ml)
    const float* __restrict__ Bbase = Ws + mhalf * PSTR + ml * 2;

    v8f acc[8] = {};

    #pragma unroll 4
    for (int k0 = 0; k0 < DCH; k0 += 4) {
        const int ka = k0 + mhalf * 2;
        float a0 = Arow[ka];
        float a1 = Arow[ka + 1];
        if (PRE) {
            a0 = fmaxf(fmaf(a0, preScale[ka],     preShift[ka]),     0.0f);
            a1 = fmaxf(fmaf(a1, preScale[ka + 1], preShift[ka + 1]), 0.0f);
        }
        v2f a;  a[0] = a0;  a[1] = a1;

        const float* __restrict__ Bk = Bbase + (k0 >> 1) * PSTR;
        #pragma unroll
        for (int t = 0; t < 8; ++t) {
            const v2f b = *reinterpret_cast<const v2f*>(Bk + t * 32); // ds_load_b64
            acc[t] = __builtin_amdgcn_wmma_f32_16x16x4_f32(
                         false, a, false, b, (short)0, acc[t], false, false);
        }
    }

    // D layout: VGPR r -> M = r + 8*mhalf, N = t*16 + ml
    #pragma unroll
    for (int t = 0; t < 8; ++t) {
        #pragma unroll
        for (int r = 0; r < 8; ++r) {
            const int row = rowBase + r + 8 * mhalf;
            if (row < nrows)
                H[(size_t)row * DCH + t * 16 + ml] = acc[t][r];
        }
    }
}

// ---------------------------------------------------------------------------
// Graph-norm helpers
// ---------------------------------------------------------------------------
__global__ void init_deg(float* __restrict__ deg, int n)
{
    const int i = blockIdx.x * blockDim.x + threadIdx.x;
    if (i < n) deg[i] = 1.0f;                        // self-loop contribution
}

__global__ void deg_accum(const int* __restrict__ dst, float* __restrict__ deg, int ne)
{
    const int e = blockIdx.x * blockDim.x + threadIdx.x;
    if (e < ne) unsafeAtomicAdd(&deg[dst[e]], 1.0f); // global_atomic_add_f32
}

__global__ void deg_to_dinv(float* __restrict__ deg, int n)
{
    const int i = blockIdx.x * blockDim.x + threadIdx.x;
    if (i < n) deg[i] = __frsqrt_rn(deg[i]);         // deg >= 1 always
}

// Fold bias+BN into a single affine: y = a*scale + shift (ReLU applied on use)
__global__ void make_affine(const float* __restrict__ b, const float* __restrict__ g,
                            const float* __restrict__ beta, const float* __restrict__ m,
                            const float* __restrict__ v,
                            float* __restrict__ scale, float* __restrict__ shift)
{
    const int d = threadIdx.x;       // 128 threads
    const float s = g[d] * __frsqrt_rn(v[d] + BN_EPS);
    scale[d] = s;
    shift[d] = (b[d] - m[d]) * s + beta[d];
}

// out[i,:] = H[i,:] * dinv[i]^2   (self-loop term; also initializes the buffer)
__global__ void selfloop_init(const float* __restrict__ H, const float* __restrict__ dinv,
                              float* __restrict__ out, int n)
{
    const int idx  = blockIdx.x * blockDim.x + threadIdx.x; // n*32 threads
    const int node = idx >> 5;
    if (node >= n) return;
    const int c = (idx & 31) << 2;
    const float w = dinv[node] * dinv[node];
    const float4 h = *reinterpret_cast<const float4*>(H + (size_t)node * DCH + c);
    float4 o;  o.x = h.x * w;  o.y = h.y * w;  o.z = h.z * w;  o.w = h.w * w;
    *reinterpret_cast<float4*>(out + (size_t)node * DCH + c) = o;
}

// One wave per edge: out[dst,:] += H[src,:] * dinv[src]*dinv[dst]  (L2 atomics)
__global__ void edge_scatter(const float* __restrict__ H,
                             const int* __restrict__ src, const int* __restrict__ dst,
                             const float* __restrict__ dinv,
                             float* __restrict__ out, int ne)
{
    const int e = blockIdx.x * (blockDim.x >> 5) + (threadIdx.x >> 5);
    if (e >= ne) return;
    const int lane = threadIdx.x & 31;
    const int s = src[e];
    const int t = dst[e];
    const float w = dinv[s] * dinv[t];
    const int c = lane << 2;
    const float4 h = *reinterpret_cast<const float4*>(H + (size_t)s * DCH + c);
    float* o = out + (size_t)t * DCH + c;
    unsafeAtomicAdd(o + 0, h.x * w);
    unsafeAtomicAdd(o + 1, h.y * w);
    unsafeAtomicAdd(o + 2, h.z * w);
    unsafeAtomicAdd(o + 3, h.w * w);
}

// Final layer epilogue: d_out = agg + b2
__global__ void add_bias(const float* __restrict__ agg, const float* __restrict__ b,
                         float* __restrict__ out, size_t total)
{
    const size_t idx = (size_t)blockIdx.x * blockDim.x + threadIdx.x;
    if (idx < total) out[idx] = agg[idx] + b[idx & (DCH - 1)];
}

// ---------------------------------------------------------------------------
extern "C" void kernel_launch(void* const* d_in, const int* in_sizes, int n_in,
                              void* d_out, int out_size, void* d_ws, size_t ws_size,
                              hipStream_t stream)
{
    const float* x     = (const float*)d_in[0];
    const int*   eidx  = (const int*)  d_in[1];   // [2, E]
    const float* eattr = (const float*)d_in[2];   // [E, 8]
    const float* W0 = (const float*)d_in[3];  const float* b0 = (const float*)d_in[4];
    const float* W1 = (const float*)d_in[5];  const float* b1 = (const float*)d_in[6];
    const float* W2 = (const float*)d_in[7];  const float* b2 = (const float*)d_in[8];
    const float* g0 = (const float*)d_in[9];  const float* be0 = (const float*)d_in[10];
    const float* m0 = (const float*)d_in[11]; const float* v0  = (const float*)d_in[12];
    const float* g1 = (const float*)d_in[13]; const float* be1 = (const float*)d_in[14];
    const float* m1 = (const float*)d_in[15]; const float* v1  = (const float*)d_in[16];

    const int N = in_sizes[0] / DCH;
    const int E = in_sizes[1] / 2;
    const int* src = eidx;
    const int* dst = eidx + E;

    // Workspace carve-up
    char* ws = (char*)d_ws;
    float* H    = (float*)ws;  ws += (size_t)N * DCH * sizeof(float);
    float* AGG  = (float*)ws;  ws += (size_t)N * DCH * sizeof(float);
    float* dinv = (float*)ws;  ws += (size_t)N * sizeof(float);
    float* sc0  = (float*)ws;  ws += DCH * sizeof(float);
    float* sh0  = (float*)ws;  ws += DCH * sizeof(float);
    float* sc1  = (float*)ws;  ws += DCH * sizeof(float);
    float* sh1  = (float*)ws;  ws += DCH * sizeof(float);

    const int gN    = (N + 255) / 256;
    const int gE    = (E + 255) / 256;
    const int gRow  = (N + 127) / 128;                 // gemm blocks
    const int gNode = (int)(((size_t)N * 32 + 255) / 256);  // selfloop_init
    const int gEdge = (E + 7) / 8;                     // 8 edges / 256-thread block

    // Degree / normalization
    init_deg   <<<gN, 256, 0, stream>>>(dinv, N);
    deg_accum  <<<gE, 256, 0, stream>>>(dst, dinv, E);
    deg_to_dinv<<<gN, 256, 0, stream>>>(dinv, N);

    // Fused bias+BN affines for layers 0 and 1
    make_affine<<<1, DCH, 0, stream>>>(b0, g0, be0, m0, v0, sc0, sh0);
    make_affine<<<1, DCH, 0, stream>>>(b1, g1, be1, m1, v1, sc1, sh1);

    // Layer 0
    gemm_wmma<false><<<gRow, 256, 0, stream>>>(x, W0, H, N, nullptr, nullptr);
    selfloop_init<<<gNode, 256, 0, stream>>>(H, dinv, AGG, N);
    edge_scatter <<<gEdge, 256, 0, stream>>>(H, src, dst, dinv, AGG, E);

    // Layer 1 (bias0+BN0+ReLU fused into A-load)
    gemm_wmma<true><<<gRow, 256, 0, stream>>>(AGG, W1, H, N, sc0, sh0);
    selfloop_init<<<gNode, 256, 0, stream>>>(H, dinv, AGG, N);
    edge_scatter <<<gEdge, 256, 0, stream>>>(H, src, dst, dinv, AGG, E);

    // Layer 2 (bias1+BN1+ReLU fused into A-load)
    gemm_wmma<true><<<gRow, 256, 0, stream>>>(AGG, W2, H, N, sc1, sh1);
    selfloop_init<<<gNode, 256, 0, stream>>>(H, dinv, AGG, N);
    edge_scatter <<<gEdge, 256, 0, stream>>>(H, src, dst, dinv, AGG, E);

    // Final bias, write node features to d_out
    const size_t total = (size_t)N * DCH;
    add_bias<<<(int)((total + 255) / 256), 256, 0, stream>>>(AGG, b2, (float*)d_out, total);

    // Second tuple element: passthrough edge_attr
    hipMemcpyAsync((float*)d_out + total, eattr, (size_t)E * 8 * sizeof(float),
                   hipMemcpyDeviceToDevice, stream);
}